// Lossfunction_24498493456944
// MI455X (gfx1250) — compile-verified
//
#include <hip/hip_runtime.h>
#include <hip/hip_bf16.h>
#include <math.h>

typedef float v2f __attribute__((ext_vector_type(2)));
typedef float v4f __attribute__((ext_vector_type(4)));
typedef float v8f __attribute__((ext_vector_type(8)));

#define ZDIM    1024
#define BATCH   2048
#define NPOS    8
#define NNEG    16
#define EPS     1e-6f
#define MARGIN  1.0f

// One block per batch element. 64 threads = 2 waves (wave32).
// wave0: Gram tile of rows {anc, x0..x14}
// wave1: Gram tile of rows {anc, x15..x23, anc-pad x6}
// where x0..x7 = pos rows, x8..x23 = neg rows of this batch.
//
// V_WMMA_F32_16X16X4_F32 computes D = A(16x4) x B(4x16) + C.
// f32 A layout: lane L<16 holds {A[L,0],A[L,1]} in 2 VGPRs, lane L+16 holds
// {A[L,2],A[L,3]}.  B mirrors this (rows k / k+2 split across lane halves),
// so for a symmetric Gram G = R * R^T the SAME register pair is passed as
// both A and B.  Each lane loads float4 row[(k0 + 4h) .. +3] (h = lane>>4)
// and issues two WMMAs per load; since A==B, the K-index permutation is
// consistent between the operands, so the partial sums are exact.
__global__ __launch_bounds__(64)
void triplet_gram_kernel(const float* __restrict__ anc,
                         const float* __restrict__ pos,
                         const float* __restrict__ neg,
                         float* __restrict__ partial)
{
    __shared__ float gram[2][16][16];   // [tile][M][N]
    __shared__ float rsum[2][16];       // per-row element sums (for eps term)
    __shared__ float dval[32];          // distances d(x_j), j = 0..23
    __shared__ float red[64];

    const int b    = blockIdx.x;
    const int tid  = threadIdx.x;
    const int wave = tid >> 5;          // tile id
    const int lane = tid & 31;
    const int r    = lane & 15;         // row within tile
    const int h    = lane >> 4;         // K-half selector

    // per-lane row pointer
    const float* rowp;
    if (r == 0) {
        rowp = anc + (size_t)b * ZDIM;
    } else if (wave == 0) {
        if (r <= 8) rowp = pos + ((size_t)b * NPOS + (r - 1)) * ZDIM;   // x0..x7
        else        rowp = neg + ((size_t)b * NNEG + (r - 9)) * ZDIM;   // x8..x14
    } else {
        if (r <= 9) rowp = neg + ((size_t)b * NNEG + (r + 6)) * ZDIM;   // x15..x23
        else        rowp = anc + (size_t)b * ZDIM;                      // padding
    }
    const v4f* p4 = (const v4f*)(rowp + h * 4);

    v8f acc = {};       // 16x16 f32 accumulator (8 VGPRs)
    float s = 0.f;      // partial row sum (this lane's K subset)

    #pragma unroll 4
    for (int k0 = 0; k0 < ZDIM; k0 += 8) {
        v4f x = p4[k0 >> 2];            // row[(k0 + 4h) .. (k0 + 4h + 3)]
        v2f a0; a0.x = x.x; a0.y = x.y;
        v2f a1; a1.x = x.z; a1.y = x.w;
        acc = __builtin_amdgcn_wmma_f32_16x16x4_f32(
                  false, a0, false, a0, (short)0, acc, false, false);
        acc = __builtin_amdgcn_wmma_f32_16x16x4_f32(
                  false, a1, false, a1, (short)0, acc, false, false);
        s += x.x + x.y + x.z + x.w;
    }

    // row sum = this lane's half + the other K-half (lane ^ 16)
    s += __shfl_xor(s, 16, 32);
    if (h == 0) rsum[wave][r] = s;

    // D layout: acc[v] = G[v + 8*h][lane & 15]
    #pragma unroll
    for (int v = 0; v < 8; ++v)
        gram[wave][v + 8 * h][r] = acc[v];

    __syncthreads();

    // distances: lane j -> x_j  (j<8: pos, j>=8: neg)
    if (tid < 24) {
        const int t = (tid < 15) ? 0 : 1;
        const int m = (tid < 15) ? tid + 1 : tid - 14;
        const float Gaa = gram[0][0][0];
        const float Gax = gram[t][0][m];
        const float Gxx = gram[t][m][m];
        const float Sa  = rsum[0][0];
        const float Sx  = rsum[t][m];
        float dsq = Gaa + Gxx - 2.0f * Gax
                  + 2.0f * EPS * (Sa - Sx)
                  + (float)ZDIM * EPS * EPS;
        dval[tid] = sqrtf(fmaxf(dsq, 0.0f));
    }
    __syncthreads();

    // 128 hinge terms over 64 threads
    float part = 0.f;
    for (int c = tid; c < NPOS * NNEG; c += 64) {
        const int p = c >> 4;
        const int n = c & 15;
        part += fmaxf(dval[p] - dval[8 + n] + MARGIN, 0.0f);
    }
    red[tid] = part;
    __syncthreads();
    #pragma unroll
    for (int off = 32; off > 0; off >>= 1) {
        if (tid < off) red[tid] += red[tid + off];
        __syncthreads();
    }
    if (tid == 0) partial[b] = red[0];
}

// deterministic final reduction: fixed strided sums + fixed LDS tree
__global__ __launch_bounds__(256)
void triplet_reduce_kernel(const float* __restrict__ partial,
                           float* __restrict__ out)
{
    __shared__ float red[256];
    float s = 0.f;
    for (int i = threadIdx.x; i < BATCH; i += 256) s += partial[i];
    red[threadIdx.x] = s;
    __syncthreads();
    #pragma unroll
    for (int off = 128; off > 0; off >>= 1) {
        if ((int)threadIdx.x < off) red[threadIdx.x] += red[threadIdx.x + off];
        __syncthreads();
    }
    if (threadIdx.x == 0)
        out[0] = red[0] * (1.0f / (float)(BATCH * NPOS * NNEG));
}

extern "C" void kernel_launch(void* const* d_in, const int* in_sizes, int n_in,
                              void* d_out, int out_size, void* d_ws, size_t ws_size,
                              hipStream_t stream)
{
    const float* anc = (const float*)d_in[0];
    const float* pos = (const float*)d_in[1];
    const float* neg = (const float*)d_in[2];
    float* partial   = (float*)d_ws;        // BATCH floats (8 KB)

    triplet_gram_kernel<<<BATCH, 64, 0, stream>>>(anc, pos, neg, partial);
    triplet_reduce_kernel<<<1, 256, 0, stream>>>(partial, (float*)d_out);
}